// FixedSeedMixtureLowRankRNN_48369921687959
// MI455X (gfx1250) — compile-verified
//
#include <hip/hip_runtime.h>
#include <math.h>

#define RANKC 2
#define ALPHAC 0.1f
#define HIDDENC 512
#define INPUTC 8
#define NUM_MIXC 2
#define NUM_SEEDSC 4
#define BATCHC 64
#define SEQC 2048
#define DC 12      // 2*RANK + INPUT
#define NSC 10     // RANK + INPUT (span columns)

typedef __attribute__((ext_vector_type(2))) float v2f;
typedef __attribute__((ext_vector_type(8))) float v8f;

// workspace layout (in floats)
#define WS_M0 0
#define WS_M1 2048
#define WS_N0 4096
#define WS_N1 6144
#define WS_I  8192                       // [s][h][i]  4*512*8
#define WS_PS 24576                      // [s][k][l]  4*10*10
#define WS_IX 32768                      // [b][t][h]  64*2048*512
#define WS_IX_COUNT (64ull * 2048ull * 512ull)

// ---------------------------------------------------------------------------
// Kernel A: per-seed prep. Computes clamped/mixed L, comb = L~ @ seeds + mean~,
// splits into m0,m1,n0,n1,I planes; computes G = span^T span (10x10), inverts
// it (Gauss-Jordan w/ partial pivoting), and stores PS = G^-1 G = P @ span.
// ---------------------------------------------------------------------------
__global__ void __launch_bounds__(128)
prep_kernel(const float* __restrict__ means, const float* __restrict__ st,
            const float* __restrict__ mw, const float* __restrict__ seeds,
            float* __restrict__ ws) {
  __shared__ float Lbar[DC][DC];
  __shared__ float mbar[DC];
  __shared__ float comb[HIDDENC][DC];
  __shared__ float G[NSC][NSC];
  __shared__ float aug[NSC][2 * NSC];
  const int s = blockIdx.x;
  const int tid = threadIdx.x;

  float w0 = fmaxf(mw[0], 1e-6f);
  float w1 = fmaxf(mw[1], 1e-6f);
  const float wsum = w0 + w1;
  w0 /= wsum; w1 /= wsum;
  const float wv[2] = {w0, w1};

  for (int idx = tid; idx < DC * DC; idx += blockDim.x) {
    const int d = idx / DC, e = idx % DC;
    float acc = 0.f;
    for (int i = 0; i < NUM_MIXC; ++i) {
      float v = st[((size_t)i * DC + d) * DC + e];
      if (e > d) v = 0.f;
      else if (e == d) v = fabsf(v - 1e-12f) + 1e-12f;
      acc += wv[i] * v;
    }
    Lbar[d][e] = acc;
  }
  for (int d = tid; d < DC; d += blockDim.x)
    mbar[d] = w0 * means[d] + w1 * means[DC + d];
  __syncthreads();

  for (int idx = tid; idx < HIDDENC * DC; idx += blockDim.x) {
    const int h = idx / DC, d = idx % DC;
    const float* se = seeds + ((size_t)s * HIDDENC + h) * DC;
    float acc = mbar[d];
    for (int e = 0; e < DC; ++e) acc += Lbar[d][e] * se[e];
    comb[h][d] = acc;
    if      (d == 0) ws[WS_M0 + s * HIDDENC + h] = acc;
    else if (d == 1) ws[WS_M1 + s * HIDDENC + h] = acc;
    else if (d == 2) ws[WS_N0 + s * HIDDENC + h] = acc;
    else if (d == 3) ws[WS_N1 + s * HIDDENC + h] = acc;
    else             ws[WS_I + ((size_t)s * HIDDENC + h) * INPUTC + (d - 4)] = acc;
  }
  __syncthreads();

  if (tid < NSC * NSC) {  // G[k][l] = sum_h span[h][k] * span[h][l]
    const int k = tid / NSC, l = tid % NSC;
    const int ck = (k < RANKC) ? k : k + RANKC;   // span cols: m(0,1), I(4..11)
    const int cl = (l < RANKC) ? l : l + RANKC;
    float acc = 0.f;
    for (int h = 0; h < HIDDENC; ++h) acc += comb[h][ck] * comb[h][cl];
    G[k][l] = acc;
  }
  __syncthreads();

  if (tid == 0) {  // tiny serial Gauss-Jordan inverse of G
    for (int r = 0; r < NSC; ++r)
      for (int c = 0; c < NSC; ++c) {
        aug[r][c] = G[r][c];
        aug[r][NSC + c] = (r == c) ? 1.f : 0.f;
      }
    for (int col = 0; col < NSC; ++col) {
      int piv = col;
      float best = fabsf(aug[col][col]);
      for (int r = col + 1; r < NSC; ++r) {
        const float a = fabsf(aug[r][col]);
        if (a > best) { best = a; piv = r; }
      }
      if (piv != col)
        for (int c = 0; c < 2 * NSC; ++c) {
          const float t = aug[col][c]; aug[col][c] = aug[piv][c]; aug[piv][c] = t;
        }
      const float inv = 1.f / aug[col][col];
      for (int c = 0; c < 2 * NSC; ++c) aug[col][c] *= inv;
      for (int r = 0; r < NSC; ++r)
        if (r != col) {
          const float f = aug[r][col];
          for (int c = 0; c < 2 * NSC; ++c) aug[r][c] -= f * aug[col][c];
        }
    }
  }
  __syncthreads();

  if (tid < NSC * NSC) {  // PS = G^-1 @ G  (== P @ span, numeric left-inverse)
    const int k = tid / NSC, l = tid % NSC;
    float acc = 0.f;
    for (int j = 0; j < NSC; ++j) acc += aug[k][NSC + j] * G[j][l];
    ws[WS_PS + ((size_t)s * NSC + k) * NSC + l] = acc;
  }
}

// ---------------------------------------------------------------------------
// Kernel B: WMMA batched GEMM  Ix[b,t,h] = sum_i x[b,t,i] * I_{seed(b)}[h,i]
// One wave per (b, 16-row t-tile); loops over 32 h-tiles; K=8 via two chained
// V_WMMA_F32_16X16X4_F32.
// ---------------------------------------------------------------------------
__global__ void __launch_bounds__(256)
ix_wmma_kernel(const float* __restrict__ x, const int* __restrict__ cur_seeds,
               float* __restrict__ ws) {
  const int gw = blockIdx.x * (blockDim.x >> 5) + (threadIdx.x >> 5);
  const int lane = threadIdx.x & 31;
  const int b = gw >> 7;                 // 64 batches
  const int t0 = (gw & 127) << 4;        // 128 t-tiles of 16
  const int s = cur_seeds[b];
  const float* Ib = ws + WS_I + (size_t)s * HIDDENC * INPUTC;
  float* Ix = ws + WS_IX;

  const int colbase = (lane >> 4) << 1;  // 0 or 2
  const int mrow = lane & 15;

  const float* xrow = x + ((size_t)b * SEQC + t0 + mrow) * INPUTC;
  v2f a0, a1;
  a0.x = xrow[colbase];     a0.y = xrow[colbase + 1];       // K chunk 0 (i=0..3)
  a1.x = xrow[4 + colbase]; a1.y = xrow[4 + colbase + 1];   // K chunk 1 (i=4..7)

  const size_t outbase = ((size_t)b * SEQC + t0 + ((lane >> 4) << 3)) * HIDDENC + mrow;
  for (int h0 = 0; h0 < HIDDENC; h0 += 16) {
    const float* Irow = Ib + (size_t)(h0 + mrow) * INPUTC;
    v2f b0, b1;
    b0.x = Irow[colbase];     b0.y = Irow[colbase + 1];
    b1.x = Irow[4 + colbase]; b1.y = Irow[4 + colbase + 1];
    v8f c = {};
    c = __builtin_amdgcn_wmma_f32_16x16x4_f32(false, a0, false, b0, (short)0, c, false, false);
    c = __builtin_amdgcn_wmma_f32_16x16x4_f32(false, a1, false, b1, (short)0, c, false, false);
    float* op = Ix + outbase + h0;
#pragma unroll
    for (int r = 0; r < 8; ++r) op[(size_t)r * HIDDENC] = c[r];
  }
}

// ---------------------------------------------------------------------------
// Kernel C: the serial RNN scan. One wave32 per batch; lane owns 16 h's.
// Rank-2 trick: J@tanh(h) = c*(m0*u0 + m1*u1), u_r via xor-shuffle reduction.
// Output o = P@h maintained as a 10-dim recursion via PS. The t-loop is
// software-pipelined: loads for step t+1 issue before the tanh/reduce chain
// of step t so L2 latency hides under the serial VALU work.
// ---------------------------------------------------------------------------
template <bool PRECOMP>
__global__ void __launch_bounds__(32)
rnn_kernel(const float* __restrict__ x, const int* __restrict__ cur_seeds,
           const float* __restrict__ ws, float* __restrict__ out) {
  const int b = blockIdx.x;
  const int lane = threadIdx.x;
  const int s = cur_seeds[b];
  const int hbase = lane * 16;
  const float cJ = 500.0f / (float)HIDDENC;

  float m0v[16], m1v[16], n0v[16], n1v[16];
#pragma unroll
  for (int j = 0; j < 16; ++j) {
    m0v[j] = ws[WS_M0 + s * HIDDENC + hbase + j];
    m1v[j] = ws[WS_M1 + s * HIDDENC + hbase + j];
    n0v[j] = ws[WS_N0 + s * HIDDENC + hbase + j];
    n1v[j] = ws[WS_N1 + s * HIDDENC + hbase + j];
  }

  float Ireg[16][8];
  if constexpr (!PRECOMP) {
#pragma unroll
    for (int j = 0; j < 16; ++j)
#pragma unroll
      for (int i = 0; i < 8; ++i)
        Ireg[j][i] = ws[WS_I + ((size_t)s * HIDDENC + hbase + j) * INPUTC + i];
  }

  float ps[10];
#pragma unroll
  for (int l = 0; l < 10; ++l)
    ps[l] = (lane < 10) ? ws[WS_PS + ((size_t)s * NSC + lane) * NSC + l] : 0.f;

  const float* Ix = ws + WS_IX + (size_t)b * SEQC * HIDDENC + hbase;
  const float* xb = x + (size_t)b * SEQC * INPUTC;
  float* ob = out + (size_t)b * SEQC * NSC;

  float h[16];
#pragma unroll
  for (int j = 0; j < 16; ++j) h[j] = 0.f;
  float o = 0.f;

  // software-pipeline prologue: data for t = 0
  float xt[8];
#pragma unroll
  for (int i = 0; i < 8; ++i) xt[i] = xb[i];
  float ixv[16];
  if constexpr (PRECOMP) {
#pragma unroll
    for (int j = 0; j < 16; ++j) ixv[j] = Ix[j];
  }

#pragma unroll 2
  for (int t = 0; t < SEQC; ++t) {
    // ---- issue next-step loads first (clamped tail index, uniform) ----
    const int tn = (t + 1 < SEQC) ? (t + 1) : t;
    float xtn[8];
#pragma unroll
    for (int i = 0; i < 8; ++i) xtn[i] = xb[(size_t)tn * INPUTC + i];
    float ixn[16];
    if constexpr (PRECOMP) {
      const float* ixp = Ix + (size_t)tn * HIDDENC;
#pragma unroll
      for (int j = 0; j < 16; ++j) ixn[j] = ixp[j];
    }

    // ---- current-step input drive ----
    float ixcur[16];
    if constexpr (PRECOMP) {
#pragma unroll
      for (int j = 0; j < 16; ++j) ixcur[j] = ixv[j];
    } else {
#pragma unroll
      for (int j = 0; j < 16; ++j) {
        float acc = 0.f;
#pragma unroll
        for (int i = 0; i < 8; ++i) acc += Ireg[j][i] * xt[i];
        ixcur[j] = acc;
      }
    }

    // ---- u_r = sum_h n_r[h] * tanh(h) ----
    float p0 = 0.f, p1 = 0.f;
#pragma unroll
    for (int j = 0; j < 16; ++j) {
      const float th = tanhf(h[j]);
      p0 += n0v[j] * th;
      p1 += n1v[j] * th;
    }
#pragma unroll
    for (int off = 16; off >= 1; off >>= 1) {
      p0 += __shfl_xor(p0, off, 32);
      p1 += __shfl_xor(p1, off, 32);
    }

    // ---- h <- 0.9 h + 0.1 * (c*(m0*u0 + m1*u1) + I x_t) ----
#pragma unroll
    for (int j = 0; j < 16; ++j)
      h[j] = (1.f - ALPHAC) * h[j] + ALPHAC * (cJ * (m0v[j] * p0 + m1v[j] * p1) + ixcur[j]);

    // ---- o_k <- 0.9 o_k + 0.1 * (c*(PS[k,0]u0 + PS[k,1]u1) + PS[k,2:].x_t) ----
    if (lane < 10) {
      float drive = cJ * (ps[0] * p0 + ps[1] * p1);
#pragma unroll
      for (int i = 0; i < 8; ++i) drive += ps[2 + i] * xt[i];
      o = (1.f - ALPHAC) * o + ALPHAC * drive;
      ob[(size_t)t * NSC + lane] = o;
    }

    // ---- rotate pipeline registers ----
#pragma unroll
    for (int i = 0; i < 8; ++i) xt[i] = xtn[i];
    if constexpr (PRECOMP) {
#pragma unroll
      for (int j = 0; j < 16; ++j) ixv[j] = ixn[j];
    }
  }
}

extern "C" void kernel_launch(void* const* d_in, const int* in_sizes, int n_in,
                              void* d_out, int out_size, void* d_ws, size_t ws_size,
                              hipStream_t stream) {
  const float* x     = (const float*)d_in[0];  // (64,2048,8)
  const float* means = (const float*)d_in[1];  // (2,12)
  const float* st    = (const float*)d_in[2];  // (2,12,12)
  const float* mw    = (const float*)d_in[3];  // (2,)
  const float* seeds = (const float*)d_in[4];  // (4,512,12)
  const int*   cur   = (const int*)d_in[5];    // (64,)
  float* ws  = (float*)d_ws;
  float* out = (float*)d_out;                  // (64,2048,10) f32

  prep_kernel<<<NUM_SEEDSC, 128, 0, stream>>>(means, st, mw, seeds, ws);

  const bool precomp =
      ws_size >= (size_t)(WS_IX + WS_IX_COUNT) * sizeof(float);
  if (precomp) {
    // 8192 waves = 64 batches x 128 t-tiles; 8 waves per 256-thread block
    ix_wmma_kernel<<<1024, 256, 0, stream>>>(x, cur, ws);
    rnn_kernel<true><<<BATCHC, 32, 0, stream>>>(x, cur, ws, out);
  } else {
    rnn_kernel<false><<<BATCHC, 32, 0, stream>>>(x, cur, ws, out);
  }
}